// AssocScan_5970004541513
// MI455X (gfx1250) — compile-verified
//
#include <hip/hip_runtime.h>

// Linear recurrence scan  out[t] = g[t]*out[t-1] + x[t]  over axis n.
// Shapes fixed by the reference: b=4, n=8192, d=1024, fp32.
// Memory-bound (384 MiB mandatory traffic, ~16.5 us @ 23.3 TB/s).
// 3-pass chunked scan: per-chunk affine summaries -> tiny carry scan ->
// re-scan with carry + streamed output.
//
// Cache policy reasoning (192 MB L2):
//  - pass 1 loads RT: the same data is re-read by pass 3, and pass 3 walks
//    blocks in reverse so it first consumes the L2-resident tail of pass 1.
//  - pass 3 loads NT: every element is read for the LAST time by exactly one
//    block -> don't let them evict the still-useful pass-1 tail.
//  - output stores NT: never re-read.

typedef __attribute__((ext_vector_type(4))) float v4f;

#define B      4
#define N      8192
#define D      1024
#define NC     128          // chunks per sequence
#define L      (N / NC)     // 64 timesteps per chunk
#define CHAINS (B * D)      // 4096 independent recurrence chains
#define PFD    28           // prefetch distance in rows (rows are 4 KiB apart)

// ---------------------------------------------------------------------------
// Pass 1: each block = one (batch, chunk); thread q owns 4 contiguous channels.
// Computes the chunk's composed affine map: G = prod(g), I = local scan tail.
// ---------------------------------------------------------------------------
__global__ __launch_bounds__(256)
void AssocScan_pass1(const float* __restrict__ gates,
                     const float* __restrict__ inputs,
                     float* __restrict__ wsG,
                     float* __restrict__ wsI) {
    const int b = blockIdx.x / NC;
    const int c = blockIdx.x % NC;
    const int q = threadIdx.x;                         // channel quad 0..255
    const size_t base = ((size_t)b * N + (size_t)c * L) * D + (size_t)q * 4;
    const float* gp = gates  + base;
    const float* xp = inputs + base;

    v4f G = {1.f, 1.f, 1.f, 1.f};
    v4f I = {0.f, 0.f, 0.f, 0.f};
#pragma unroll 8
    for (int t = 0; t < L; ++t) {
        v4f g = *(const v4f*)(gp + (size_t)t * D);     // global_load_b128 (RT)
        v4f x = *(const v4f*)(xp + (size_t)t * D);
        __builtin_prefetch(gp + (size_t)(t + PFD) * D, 0, 3); // global_prefetch_b8
        __builtin_prefetch(xp + (size_t)(t + PFD) * D, 0, 3);
        G = G * g;                                     // compose affine maps
        I = g * I + x;                                 // v_fmac_f32
    }
    const size_t s = (size_t)c * CHAINS + (size_t)b * D + (size_t)q * 4;
    *(v4f*)(wsG + s) = G;
    *(v4f*)(wsI + s) = I;
}

// ---------------------------------------------------------------------------
// Pass 2: one thread per chain scans its NC chunk summaries sequentially and
// overwrites wsI[c] with the *exclusive* carry-in value for chunk c.
// (ws is fully rewritten by pass 1 each call -> deterministic under replay.)
// ---------------------------------------------------------------------------
__global__ __launch_bounds__(256)
void AssocScan_pass2(const float* __restrict__ wsG,
                     float* __restrict__ wsI) {
    const int chain = blockIdx.x * 256 + threadIdx.x;  // 0..4095
    float v = 0.f;                                     // scan starts from 0
#pragma unroll 4
    for (int c = 0; c < NC; ++c) {
        const size_t s = (size_t)c * CHAINS + (size_t)chain;
        const float Gc = wsG[s];
        const float Ic = wsI[s];
        wsI[s] = v;                                    // carry-in for chunk c
        v = Gc * v + Ic;                               // advance composed map
    }
}

// ---------------------------------------------------------------------------
// Pass 3: re-scan each chunk seeded with its carry and stream the outputs.
// Blocks walk in REVERSE of pass 1's launch order so the ~192 MB of pass-1
// data still resident in L2 (freshest tail) is consumed first.
// ---------------------------------------------------------------------------
__global__ __launch_bounds__(256)
void AssocScan_pass3(const float* __restrict__ gates,
                     const float* __restrict__ inputs,
                     const float* __restrict__ carry,   // = wsI after pass 2
                     float* __restrict__ out) {
    const int rb = (int)(gridDim.x - 1u - blockIdx.x);  // reverse order
    const int b = rb / NC;
    const int c = rb % NC;
    const int q = threadIdx.x;
    const size_t base = ((size_t)b * N + (size_t)c * L) * D + (size_t)q * 4;
    const float* gp = gates  + base;
    const float* xp = inputs + base;
    float*       op = out    + base;

    v4f acc = *(const v4f*)(carry + (size_t)c * CHAINS + (size_t)b * D + (size_t)q * 4);
#pragma unroll 8
    for (int t = 0; t < L; ++t) {
        // last-use reads: non-temporal so they don't evict useful L2 lines
        v4f g = __builtin_nontemporal_load((const v4f*)(gp + (size_t)t * D));
        v4f x = __builtin_nontemporal_load((const v4f*)(xp + (size_t)t * D));
        __builtin_prefetch(gp + (size_t)(t + PFD) * D, 0, 3);
        __builtin_prefetch(xp + (size_t)(t + PFD) * D, 0, 3);
        acc = g * acc + x;                              // the recurrence itself
        __builtin_nontemporal_store(acc, (v4f*)(op + (size_t)t * D)); // TH_NT
    }
}

// ---------------------------------------------------------------------------
extern "C" void kernel_launch(void* const* d_in, const int* in_sizes, int n_in,
                              void* d_out, int out_size, void* d_ws, size_t ws_size,
                              hipStream_t stream) {
    const float* gates  = (const float*)d_in[0];
    const float* inputs = (const float*)d_in[1];
    float*       out    = (float*)d_out;

    float* wsG = (float*)d_ws;                         // NC*CHAINS floats (2 MiB)
    float* wsI = wsG + (size_t)NC * CHAINS;            // NC*CHAINS floats (2 MiB)

    AssocScan_pass1<<<dim3(B * NC), dim3(256), 0, stream>>>(gates, inputs, wsG, wsI);
    AssocScan_pass2<<<dim3(CHAINS / 256), dim3(256), 0, stream>>>(wsG, wsI);
    AssocScan_pass3<<<dim3(B * NC), dim3(256), 0, stream>>>(gates, inputs, wsI, out);
}